// MultiHeadAttention_82377472737444
// MI455X (gfx1250) — compile-verified
//
#include <hip/hip_runtime.h>
#include <math.h>

// Transformer-XL style relative multi-head attention, MI455X / gfx1250.
// GEMMs on V_WMMA_F32_16X16X4_F32; weight tiles staged to LDS with the
// CDNA5 async global->LDS path (ASYNCcnt) when available.

#define SEQ   2048
#define DM    512
#define NH    8
#define HD    64
#define NB    2
#define RP_S  (SEQ + 32)   // r projection padded with 32 zero rows per head

typedef __attribute__((ext_vector_type(2))) float v2f;
typedef __attribute__((ext_vector_type(4))) float v4f;
typedef __attribute__((ext_vector_type(8))) float v8f;
typedef __attribute__((ext_vector_type(4))) int   v4i;

#if defined(__has_builtin)
#if __has_builtin(__builtin_amdgcn_global_load_async_to_lds_b128)
#define HAVE_ASYNC_LDS 1
#endif
#endif

#ifdef HAVE_ASYNC_LDS
#if defined(__has_builtin) && __has_builtin(__builtin_amdgcn_s_wait_asynccnt)
#define WAIT_ASYNC(n) __builtin_amdgcn_s_wait_asynccnt(n)
#else
#define WAIT_ASYNC(n) asm volatile("s_wait_asynccnt %0" ::"i"(n) : "memory")
#endif
#else
#define WAIT_ASYNC(n) ((void)0)
#endif

// Address-space qualified int4 pointers for the async-to-LDS builtin.
typedef __attribute__((address_space(1))) v4i* gv4i_p;
typedef __attribute__((address_space(3))) v4i* lv4i_p;

__device__ __forceinline__ gv4i_p to_gbl_v4(const void* p) {
  return (gv4i_p)(unsigned long long)p;
}
__device__ __forceinline__ lv4i_p to_lds_v4(const void* p) {
  // generic LDS pointer -> LDS offset (low 32 bits, per aperture mapping)
  return (lv4i_p)(unsigned int)(unsigned long long)p;
}

__device__ __forceinline__ v8f wmma_f32(v2f a, v2f b, v8f c) {
  // D = A(16x4,f32) * B(4x16,f32) + C(16x16,f32)
  return __builtin_amdgcn_wmma_f32_16x16x4_f32(false, a, false, b, (short)0, c,
                                               false, false);
}

// ---------------------------------------------------------------------------
// Projection GEMM: Out[m,n] = sum_d X[m,d] * W[n,d]   (X:[M,512], W:[512,512])
// Block = 4 waves covering a 64(M) x 64(N) tile; all waves share one 64-row
// W strip staged in LDS via async load-to-LDS, double buffered in 32-wide
// k-chunks.  layout 0: Out as [B,H,S,E];  layout 1: Out as [H, RP_S, E].
// ---------------------------------------------------------------------------
#define WSTRIDE 36                 // padded LDS row stride (floats)
#define WCHUNK  32                 // k-floats per staged chunk
#define NCHUNK  (DM / WCHUNK)      // 16 chunks

__device__ __forceinline__ void stage_w_chunk(const float* __restrict__ W,
                                              int n0, int kc,
                                              float* __restrict__ dst,
                                              int tid) {
#pragma unroll
  for (int q = 0; q < 4; ++q) {
    int v = tid + 128 * q;                 // 512 vec4 transfers per chunk
    int row = v >> 3;                      // 8 vec4 per 32-float row
    int c4 = v & 7;
    const float* gp = W + (size_t)(n0 + row) * DM + kc + c4 * 4;
    float* lp = dst + row * WSTRIDE + c4 * 4;
#ifdef HAVE_ASYNC_LDS
    __builtin_amdgcn_global_load_async_to_lds_b128(to_gbl_v4(gp), to_lds_v4(lp),
                                                   0, 0);
#else
    *(v4f*)lp = *(const v4f*)gp;
#endif
  }
}

__global__ void proj_gemm_kernel(const float* __restrict__ X,
                                 const float* __restrict__ W,
                                 float* __restrict__ Out, int M, int layout) {
  __shared__ float wbuf[2][64 * WSTRIDE];  // 18 KB double buffer
  int bm = blockIdx.x >> 3;                // DM/64 == 8 column strips
  int tn = blockIdx.x & 7;
  int tid = threadIdx.x;
  int wv = tid >> 5;
  int lane = tid & 31;
  int lm = lane & 15;
  int half = lane >> 4;
  int kko = 2 * half;
  int m0 = bm * 64 + wv * 16;
  int n0 = tn * 64;
  (void)M;

  stage_w_chunk(W, n0, 0, &wbuf[0][0], tid);

  v8f acc0 = {}, acc1 = {}, acc2 = {}, acc3 = {};
  const float* xrow = X + (size_t)(m0 + lm) * DM;

  for (int c = 0; c < NCHUNK; ++c) {
    int kc = c * WCHUNK;
    int buf = c & 1;
    if (c + 1 < NCHUNK) {
      stage_w_chunk(W, n0, kc + WCHUNK, &wbuf[buf ^ 1][0], tid);
      WAIT_ASYNC(4);                       // retire chunk c (in-order)
    } else {
      WAIT_ASYNC(0);
    }
    __syncthreads();
    const float* wb = &wbuf[buf][0];
#pragma unroll
    for (int ks = 0; ks < WCHUNK / 4; ++ks) {
      int kk = kc + ks * 4 + kko;
      int lo = ks * 4 + kko;
      v2f a = *(const v2f*)(xrow + kk);
      acc0 = wmma_f32(a, *(const v2f*)(wb + (0  + lm) * WSTRIDE + lo), acc0);
      acc1 = wmma_f32(a, *(const v2f*)(wb + (16 + lm) * WSTRIDE + lo), acc1);
      acc2 = wmma_f32(a, *(const v2f*)(wb + (32 + lm) * WSTRIDE + lo), acc2);
      acc3 = wmma_f32(a, *(const v2f*)(wb + (48 + lm) * WSTRIDE + lo), acc3);
    }
    __syncthreads();                       // protect buffer before re-stage
  }

#pragma unroll
  for (int t = 0; t < 4; ++t) {
    v8f acc = (t == 0) ? acc0 : (t == 1) ? acc1 : (t == 2) ? acc2 : acc3;
    int n = n0 + t * 16 + lm;
#pragma unroll
    for (int g = 0; g < 8; ++g) {
      int m = m0 + g + 8 * half;
      size_t dst;
      if (layout == 0) {
        int b = m >> 11;                   // S == 2048
        int s = m & (SEQ - 1);
        int h = n >> 6;                    // E == 64
        int e = n & 63;
        dst = (((size_t)(b * NH + h) * SEQ) + s) * HD + e;
      } else {
        int h = n >> 6;
        int e = n & 63;
        dst = ((size_t)h * RP_S + m) * HD + e;
      }
      Out[dst] = acc[g];
    }
  }
}

// Zero the 32 pad rows per head of the r projection.
__global__ void rp_pad_zero_kernel(float* __restrict__ Rp) {
  int idx = blockIdx.x * blockDim.x + threadIdx.x;  // NH*32*HD = 16384
  int h = idx >> 11;
  int rem = idx & 2047;
  int r = rem >> 6;
  int e = rem & 63;
  Rp[((size_t)h * RP_S + SEQ + r) * HD + e] = 0.0f;
}

// ---------------------------------------------------------------------------
// Attention: one wave (blockDim=32) per (b,h,16-row strip).
// Pass 1: score strip [16 x SEQ] into LDS via WMMA (ac + banded bd tiles for
//         the circulant rel-shift).  A-fragments (q+u, q+t) hoisted.
// Stats:  row max / inv-sum via two-lane split + shfl_xor(16).
// Pass 2: att = softmax(S) @ V via WMMA, exp applied on A-frag load.
// ---------------------------------------------------------------------------
__global__ void attn_kernel(const float* __restrict__ Q,
                            const float* __restrict__ K,
                            const float* __restrict__ V,
                            const float* __restrict__ Rp,
                            const float* __restrict__ u,
                            const float* __restrict__ t,
                            float* __restrict__ att) {
  extern __shared__ float sm[];
  float* sc    = sm;                  // 16 * SEQ
  float* bdb   = sm + 16 * SEQ;       // 16 * 32 (bd gather band)
  float* stats = bdb + 16 * 32;       // 32 (max, inv-sum)

  int bid = blockIdx.x;
  int it = bid & 127;                 // SEQ/16 = 128 strips
  int h  = (bid >> 7) & 7;
  int b  = bid >> 10;
  int i0 = it << 4;

  int lane = threadIdx.x;
  int lm = lane & 15;
  int half = lane >> 4;
  int kko = 2 * half;

  const float* Qp = Q + ((size_t)(b * NH + h) * SEQ) * HD;
  const float* Kp = K + ((size_t)(b * NH + h) * SEQ) * HD;
  const float* Vp = V + ((size_t)(b * NH + h) * SEQ) * HD;
  const float* Rh = Rp + (size_t)h * RP_S * HD;
  const float* uh = u + h * HD;
  const float* th = t + h * HD;
  const float* qrow = Qp + (size_t)(i0 + lm) * HD;
  int ncol = i0 + 16;

  // Hoisted A-fragments: invariant across all jt tiles of this strip.
  v2f aa[16], ap[16];
#pragma unroll
  for (int ks = 0; ks < 16; ++ks) {
    int kk = ks * 4 + kko;
    float q0 = qrow[kk], q1 = qrow[kk + 1];
    aa[ks] = (v2f){q0 + uh[kk], q1 + uh[kk + 1]};   // content query
    ap[ks] = (v2f){q0 + th[kk], q1 + th[kk + 1]};   // position query
  }

  // ----- pass 1: masked/scaled scores into LDS -----
  for (int jt = 0; jt <= it; ++jt) {
    int j0 = jt << 4;
    int base = (SEQ - 1) - i0 + j0;   // jj for (di=0,dj=0)
    int jjs = base - 15;              // band start (>= 0)

    if (jt < it) {                    // prefetch next tile's operands
      __builtin_prefetch(Kp + (size_t)(j0 + 16 + lm) * HD, 0, 1);
      __builtin_prefetch(Rh + (size_t)(jjs + 32 + lm) * HD, 0, 1);
    }

    v8f ac = {}, bd0 = {}, bd1 = {};
#pragma unroll
    for (int ks = 0; ks < 16; ++ks) {
      int kk = ks * 4 + kko;
      v2f bk  = *(const v2f*)(Kp + (size_t)(j0 + lm) * HD + kk);
      v2f br0 = *(const v2f*)(Rh + (size_t)(jjs + lm) * HD + kk);
      v2f br1 = *(const v2f*)(Rh + (size_t)(jjs + 16 + lm) * HD + kk);
      ac  = wmma_f32(aa[ks], bk,  ac);
      bd0 = wmma_f32(ap[ks], br0, bd0);
      bd1 = wmma_f32(ap[ks], br1, bd1);
    }
    // stage bd band (16 rows x 32 jj-cols) for the Toeplitz gather
#pragma unroll
    for (int g = 0; g < 8; ++g) {
      int di = g + 8 * half;
      bdb[di * 32 + lm]      = bd0[g];
      bdb[di * 32 + 16 + lm] = bd1[g];
    }
    __syncthreads();
#pragma unroll
    for (int g = 0; g < 8; ++g) {
      int di = g + 8 * half;
      int i = i0 + di;
      int j = j0 + lm;
      // bd_shift[i,j] = bdU[i][base + (dj - di)] = bdb[di][15 + dj - di]
      float val = (ac[g] + bdb[di * 32 + 15 + lm - di]) * 0.125f; // /sqrt(64)
      if (j > i || val == 0.0f) val = -INFINITY;  // tril + exact-zero quirk
      sc[di * SEQ + j] = val;
    }
    __syncthreads();
  }

  // ----- row statistics (2 lanes per row) -----
  {
    int row = lm;
    float mx = -INFINITY;
    for (int j = half; j < ncol; j += 2) mx = fmaxf(mx, sc[row * SEQ + j]);
    mx = fmaxf(mx, __shfl_xor(mx, 16, 32));
    float ssum = 0.0f;
    for (int j = half; j < ncol; j += 2) ssum += __expf(sc[row * SEQ + j] - mx);
    ssum += __shfl_xor(ssum, 16, 32);
    if (half == 0) {
      stats[row] = mx;
      stats[16 + row] = 1.0f / ssum;
    }
  }
  __syncthreads();

  // ----- pass 2: att = P @ V (exp applied on A-frag load) -----
  float mxl  = stats[lm];
  float invl = stats[16 + lm];
  v8f o0 = {}, o1 = {}, o2 = {}, o3 = {};
  for (int k0 = 0; k0 < ncol; k0 += 4) {
    int kk = k0 + kko;
    const float* vr0 = Vp + (size_t)kk * HD;
    const float* vr1 = Vp + (size_t)(kk + 1) * HD;
    __builtin_prefetch(vr0 + 8 * HD + lm, 0, 1);
    v2f a = {__expf(sc[lm * SEQ + kk] - mxl) * invl,
             __expf(sc[lm * SEQ + kk + 1] - mxl) * invl};
    v2f b0 = {vr0[lm],      vr1[lm]};
    v2f b1 = {vr0[16 + lm], vr1[16 + lm]};
    v2f b2 = {vr0[32 + lm], vr1[32 + lm]};
    v2f b3 = {vr0[48 + lm], vr1[48 + lm]};
    o0 = wmma_f32(a, b0, o0);
    o1 = wmma_f32(a, b1, o1);
    o2 = wmma_f32(a, b2, o2);
    o3 = wmma_f32(a, b3, o3);
  }
  // store att as [B, S, H*E] for the output projection
#pragma unroll
  for (int g = 0; g < 8; ++g) {
    int di = g + 8 * half;
    int row = i0 + di;
    float* dst = att + ((size_t)b * SEQ + row) * DM + h * HD;
    dst[lm]      = o0[g];
    dst[16 + lm] = o1[g];
    dst[32 + lm] = o2[g];
    dst[48 + lm] = o3[g];
  }
}

// ---------------------------------------------------------------------------
// Output projection + residual + LayerNorm. One block (4 waves) per 16 rows.
// ---------------------------------------------------------------------------
__global__ void outln_kernel(const float* __restrict__ att,
                             const float* __restrict__ Wo,
                             const float* __restrict__ x,
                             const float* __restrict__ gamma,
                             const float* __restrict__ beta,
                             float* __restrict__ out) {
  __shared__ float yb[16 * DM];
  __shared__ float psum[128], psq[128];
  __shared__ float smu[16], srs[16];

  int m0 = blockIdx.x * 16;
  int tid = threadIdx.x;
  int wv = tid >> 5;
  int lane = tid & 31;
  int lm = lane & 15;
  int half = lane >> 4;
  int kko = 2 * half;
  int n0 = wv * 128;

  v8f zero = {};
  v8f acc[8];
#pragma unroll
  for (int t = 0; t < 8; ++t) acc[t] = zero;

  const float* arow = att + (size_t)(m0 + lm) * DM;
  for (int k0 = 0; k0 < DM; k0 += 4) {
    int kk = k0 + kko;
    v2f a = *(const v2f*)(arow + kk);
#pragma unroll
    for (int t = 0; t < 8; ++t) {
      v2f bf = *(const v2f*)(Wo + (size_t)(n0 + t * 16 + lm) * DM + kk);
      acc[t] = wmma_f32(a, bf, acc[t]);
    }
  }
#pragma unroll
  for (int t = 0; t < 8; ++t) {
    int n = n0 + t * 16 + lm;
#pragma unroll
    for (int g = 0; g < 8; ++g) {
      int di = g + 8 * half;
      yb[di * DM + n] = acc[t][g] + x[(size_t)(m0 + di) * DM + n];
    }
  }
  __syncthreads();

  int row = tid >> 3;       // 16 rows x 8 partials
  int part = tid & 7;
  float s = 0.0f, sq = 0.0f;
  for (int j = part * 64; j < part * 64 + 64; ++j) {
    float v = yb[row * DM + j];
    s += v;
    sq += v * v;
  }
  psum[tid] = s;
  psq[tid] = sq;
  __syncthreads();
  if (tid < 16) {
    float ts = 0.0f, tq = 0.0f;
    for (int p = 0; p < 8; ++p) { ts += psum[tid * 8 + p]; tq += psq[tid * 8 + p]; }
    float mu = ts * (1.0f / DM);
    float var = tq * (1.0f / DM) - mu * mu;
    smu[tid] = mu;
    srs[tid] = rsqrtf(var + 1e-5f);
  }
  __syncthreads();
  float mu = smu[row], rs = srs[row];
  for (int kk = 0; kk < 64; ++kk) {
    int j = part + kk * 8;
    out[(size_t)(m0 + row) * DM + j] =
        (yb[row * DM + j] - mu) * rs * gamma[j] + beta[j];
  }
}

// ---------------------------------------------------------------------------
extern "C" void kernel_launch(void* const* d_in, const int* in_sizes, int n_in,
                              void* d_out, int out_size, void* d_ws,
                              size_t ws_size, hipStream_t stream) {
  const float* x     = (const float*)d_in[0];
  const float* R     = (const float*)d_in[1];
  const float* u     = (const float*)d_in[2];
  const float* t     = (const float*)d_in[3];
  const float* Wq    = (const float*)d_in[4];
  const float* Wk    = (const float*)d_in[5];
  const float* Wv    = (const float*)d_in[6];
  const float* Wr    = (const float*)d_in[7];
  const float* Wo    = (const float*)d_in[8];
  const float* gamma = (const float*)d_in[9];
  const float* beta  = (const float*)d_in[10];
  float* out = (float*)d_out;

  const size_t QSZ  = (size_t)NB * NH * SEQ * HD;   // 2,097,152
  const size_t RPSZ = (size_t)NH * RP_S * HD;       // 1,064,960
  float* ws  = (float*)d_ws;
  float* Qb  = ws;
  float* Kb  = Qb + QSZ;
  float* Vb  = Kb + QSZ;
  float* Rpb = Vb + QSZ;
  float* Att = Rpb + RPSZ;

  // projections: one block per 64x64 output tile
  int blkQKV = (NB * SEQ / 64) * (DM / 64);         // 512
  int blkR   = (SEQ / 64) * (DM / 64);              // 256
  proj_gemm_kernel<<<blkQKV, 128, 0, stream>>>(x, Wq, Qb, NB * SEQ, 0);
  proj_gemm_kernel<<<blkQKV, 128, 0, stream>>>(x, Wk, Kb, NB * SEQ, 0);
  proj_gemm_kernel<<<blkQKV, 128, 0, stream>>>(x, Wv, Vb, NB * SEQ, 0);
  proj_gemm_kernel<<<blkR,   128, 0, stream>>>(R, Wr, Rpb, SEQ, 1);
  rp_pad_zero_kernel<<<(NH * 32 * HD) / 256, 256, 0, stream>>>(Rpb);

  // attention: one wave per (b,h,strip); dynamic LDS for the score strip
  size_t attn_lds = (size_t)(16 * SEQ + 16 * 32 + 32) * sizeof(float);
  attn_kernel<<<NB * NH * (SEQ / 16), 32, attn_lds, stream>>>(
      Qb, Kb, Vb, Rpb, u, t, Att);

  // output projection + residual + layernorm
  outln_kernel<<<(NB * SEQ) / 16, 128, 0, stream>>>(Att, Wo, x, gamma, beta,
                                                    out);
}